// DifferentiableRBFLayer_47158740910468
// MI455X (gfx1250) — compile-verified
//
#include <hip/hip_runtime.h>
#include <hip/hip_bf16.h>

typedef float v2f __attribute__((ext_vector_type(2)));
typedef float v8f __attribute__((ext_vector_type(8)));

#define TPS_EPS 1e-6f
#define LN2F    0.69314718055994530942f

// Problem shape from the reference
#define BATCH 4
#define MCP   256      // control points
#define NDN   32768    // dense points

#define WAVES_PER_BLOCK 8
#define TPB (32 * WAVES_PER_BLOCK)        // 256 threads
#define TILE 16                           // dense points per wave
#define PER_BLOCK (WAVES_PER_BLOCK * TILE) // 128 dense points per block

__global__ __launch_bounds__(TPB)
void tps_rbf_kernel(const float* __restrict__ disp,    // (B, M, 3)
                    const float* __restrict__ cp,      // (B, M, 3)
                    const float* __restrict__ dense,   // (B, N, 3)
                    float* __restrict__ out)           // (B, N, 3)
{
    // A-operand source: per control point m, [-2cx, -2cy, -2cz, |c|^2]
    __shared__ float sAcv[MCP][4];
    // disp padded to 4 floats for aligned LDS loads
    __shared__ float sDisp[MCP][4];

    const int t = threadIdx.x;
    const int blocksPerBatch = NDN / PER_BLOCK;           // 256
    const int b = blockIdx.x / blocksPerBatch;
    const int tileBlock = blockIdx.x % blocksPerBatch;

    // Cooperative LDS build: 256 threads <-> 256 control points
    {
        const float* c3 = cp + ((size_t)b * MCP + t) * 3;
        float cx = c3[0], cy = c3[1], cz = c3[2];
        sAcv[t][0] = -2.0f * cx;
        sAcv[t][1] = -2.0f * cy;
        sAcv[t][2] = -2.0f * cz;
        sAcv[t][3] = cx * cx + cy * cy + cz * cz;
        const float* d3 = disp + ((size_t)b * MCP + t) * 3;
        sDisp[t][0] = d3[0];
        sDisp[t][1] = d3[1];
        sDisp[t][2] = d3[2];
        sDisp[t][3] = 0.0f;
    }
    __syncthreads();

    const int wave = t >> 5;
    const int lane = t & 31;
    const int half = lane >> 4;   // 0: lanes 0-15, 1: lanes 16-31
    const int l16  = lane & 15;

    // This lane's dense point (D-matrix column n = lane%16)
    const int n = tileBlock * PER_BLOCK + wave * TILE + l16;
    const float* p3 = dense + ((size_t)b * NDN + n) * 3;
    const float px = p3[0], py = p3[1], pz = p3[2];
    const float xn2 = px * px + py * py + pz * pz;

    // B operand 4x16 f32: VGPR0 holds {K0 | K2}, VGPR1 holds {K1 | K3}
    // Column vector per dense point: [x, y, z, 1]
    v2f bmat;
    if (half == 0) { bmat[0] = px; bmat[1] = py;   }   // K0, K1
    else           { bmat[0] = pz; bmat[1] = 1.0f; }   // K2, K3

    float acc0 = 0.0f, acc1 = 0.0f, acc2 = 0.0f;

    for (int g = 0; g < MCP / 16; ++g) {
        // A operand 16x4 f32: lane holds row m = g*16 + lane%16;
        // VGPR0 = {K0 | K2}, VGPR1 = {K1 | K3} selected by lane half.
        const int m0 = g * 16 + l16;
        v2f amat;
        amat[0] = sAcv[m0][2 * half];
        amat[1] = sAcv[m0][2 * half + 1];

        v8f cz8 = {};
        // D(16x16) = A(16x4) x B(4x16): D[m][n] = |c_m|^2 - 2 c_m . x_n
        v8f d = __builtin_amdgcn_wmma_f32_16x16x4_f32(
            false, amat, false, bmat, (short)0, cz8, false, false);

        // D layout: VGPR j, lane half h -> row m = g*16 + j + 8h, col n = lane%16
#pragma unroll
        for (int j = 0; j < 8; ++j) {
            float d2 = fmaxf(d[j] + xn2, 0.0f);       // squared distance
            float r  = __builtin_amdgcn_sqrtf(d2);    // raw v_sqrt_f32
            // phi = r^2 * ln(r + eps) = (d2 * ln2) * log2(r + eps)
            float phi = (d2 * LN2F) * __builtin_amdgcn_logf(r + TPS_EPS);
            int m = g * 16 + j + 8 * half;
            acc0 = fmaf(phi, sDisp[m][0], acc0);
            acc1 = fmaf(phi, sDisp[m][1], acc1);
            acc2 = fmaf(phi, sDisp[m][2], acc2);
        }
    }

    // Lanes L and L^16 hold the two row-half partial sums for the same dense n
    acc0 += __shfl_xor(acc0, 16, 32);
    acc1 += __shfl_xor(acc1, 16, 32);
    acc2 += __shfl_xor(acc2, 16, 32);

    if (half == 0) {
        float* o3 = out + ((size_t)b * NDN + n) * 3;
        o3[0] = acc0;
        o3[1] = acc1;
        o3[2] = acc2;
    }
}

extern "C" void kernel_launch(void* const* d_in, const int* in_sizes, int n_in,
                              void* d_out, int out_size, void* d_ws, size_t ws_size,
                              hipStream_t stream) {
    const float* disp  = (const float*)d_in[0];  // sparse_disp   (B, M, 3)
    const float* cp    = (const float*)d_in[1];  // original_cp   (B, M, 3)
    const float* dense = (const float*)d_in[2];  // original_dense(B, N, 3)
    float* out = (float*)d_out;                  // (B, N, 3)

    dim3 grid(BATCH * (NDN / PER_BLOCK));        // 4 * 256 = 1024 blocks
    tps_rbf_kernel<<<grid, TPB, 0, stream>>>(disp, cp, dense, out);
}